// FirstEdgeConvBlock_86268713107568
// MI455X (gfx1250) — compile-verified
//
#include <hip/hip_runtime.h>

typedef float v2f __attribute__((ext_vector_type(2)));
typedef float v8f __attribute__((ext_vector_type(8)));
typedef unsigned int v4u __attribute__((ext_vector_type(4)));
typedef int v4i __attribute__((ext_vector_type(4)));
typedef int v8i __attribute__((ext_vector_type(8)));

#define BATCH 4
#define NPTS  8192
#define KNN   20
#define OUTC  64
#define INC   10
#define BN_EPS 1e-5f
#define CHUNK 4096   // points staged in LDS per TDM pass (64 KB of float4)

// ---------------- workspace layout (float offsets) ----------------
// packed (x,y,z,|c|^2) : BATCH*NPTS*4            = 131072 floats @ 0
// knn idx (int)        : BATCH*NPTS*KNN          = 655360 ints   @ 131072
// hmax                 : BATCH*NPTS*OUTC         = 2097152       @ 786432
// hmin                 : BATCH*NPTS*OUTC         = 2097152       @ 2883584
// stats: sum[64] sq[64] scale[64] shift[64]      = 256           @ 4980736
#define WS_PACKED 0
#define WS_IDX    131072
#define WS_HMAX   786432
#define WS_HMIN   2883584
#define WS_STATS  4980736

static __device__ __forceinline__ v8f wmma_f32_16x16x4(v2f a, v2f b, v8f c) {
  return __builtin_amdgcn_wmma_f32_16x16x4_f32(false, a, false, b, (short)0, c,
                                               false, false);
}

// Tensor Data Mover: 1-D copy of nElem 4-byte elements, global -> LDS.
// D# built per CDNA5 ISA 8.3/8.4: group0 {count=1, lds_addr, global_addr[56:0],
// type=2}, group1 {data_size=4B, tensor_dim0=tile_dim0=nElem, dims1=1}.
static __device__ __forceinline__ void tdm_load_1d(const void* gaddr,
                                                   unsigned lds_off,
                                                   unsigned nElem) {
  unsigned long long ga = (unsigned long long)(size_t)gaddr;
  v4u g0;
  g0[0] = 1u;                                            // count=1, user mode
  g0[1] = lds_off;                                       // lds_addr (bytes)
  g0[2] = (unsigned)ga;                                  // global_addr[31:0]
  g0[3] = (unsigned)((ga >> 32) & 0x01FFFFFFu) | (2u << 30);  // [56:32], type=2
  v8i g1;
  g1[0] = (int)(2u << 16);                  // workgroup_mask=0, data_size=4B
  g1[1] = (int)((nElem & 0xFFFFu) << 16);   // tensor_dim0[15:0]   @bit48
  g1[2] = (int)((nElem >> 16) | (1u << 16));// tensor_dim0[31:16], tensor_dim1=1
  g1[3] = (int)((nElem & 0xFFFFu) << 16);   // tile_dim0           @bit112
  g1[4] = 1;                                // tile_dim1=1, tile_dim2=0
  g1[5] = (int)nElem;                       // tensor_dim0_stride  @bit160
  g1[6] = 0;
  g1[7] = 0;
  v4i g2 = {0, 0, 0, 0};
  v4i g3 = {0, 0, 0, 0};
#if defined(__clang_major__) && __clang_major__ >= 23
  v8i g4 = {0, 0, 0, 0, 0, 0, 0, 0};
  __builtin_amdgcn_tensor_load_to_lds(g0, g1, g2, g3, g4, 0);
#else
  __builtin_amdgcn_tensor_load_to_lds(g0, g1, g2, g3, 0);
#endif
}

// -------- kernel 0: pack coords + squared norm; zero BN accumulators --------
__global__ __launch_bounds__(256) void prep_kernel(const float* __restrict__ pts,
                                                   float4* __restrict__ packed,
                                                   float* __restrict__ stats) {
  int i = blockIdx.x * 256 + threadIdx.x;   // over BATCH*NPTS
  if (i < 128) stats[i] = 0.f;              // sum[64] + sumsq[64]
  if (i >= BATCH * NPTS) return;
  const float* p = pts + (size_t)i * 5;
  float x = p[0], y = p[1], z = p[2];
  packed[i] = make_float4(x, y, z, x * x + y * y + z * z);
}

// -------- kernel 1: fused distance (WMMA f32 16x16x4) + top-20 selection ----
// Block = 8 waves, 128 queries of one batch. Candidates staged into LDS by the
// Tensor Data Mover in two 64KB passes; every wave scans the shared copy.
// Key tile D[m][q] = sq[m] - 2*dot(c_m, c_q):
//   A[m] = (x,y,z,sq_m)  (candidates),  B[:,q] = (-2qx,-2qy,-2qz, 1).
__global__ __launch_bounds__(256) void knn_kernel(const float4* __restrict__ P,
                                                  int* __restrict__ idxOut) {
  __shared__ float4 sP[CHUNK];              // 64 KB; reused as merge scratch
  const int lane = threadIdx.x & 31;
  const int wid  = threadIdx.x >> 5;
  const int b    = blockIdx.x / (NPTS / 128);
  const int q0   = (blockIdx.x % (NPTS / 128)) * 128 + wid * 16;
  const float4* Pb = P + (size_t)b * NPTS;

  const int col  = lane & 15;
  const int half = lane >> 4;               // 0: K={0,1}, 1: K={2,3}
  const int rowBase = half ? 8 : 0;         // C-layout M offset

  // B operand: query columns
  float4 q = Pb[q0 + col];
  v2f Bop;
  Bop[0] = half ? (-2.f * q.z) : (-2.f * q.x);
  Bop[1] = half ? 1.f          : (-2.f * q.y);

  float key[KNN];
  int   idx[KNN];
#pragma unroll
  for (int i = 0; i < KNN; ++i) { key[i] = 3.4e38f; idx[i] = 0; }

  const unsigned ldsOff = (unsigned)(size_t)&sP[0];  // LDS byte offset
  for (int h = 0; h < 2; ++h) {
    if (wid == 0) {
      tdm_load_1d(Pb + h * CHUNK, ldsOff, CHUNK * 4 /* 4B elems */);
      __builtin_amdgcn_s_wait_tensorcnt(0);
    } else if (wid == 1) {
      // warm GL2 for the next TDM pass while pass h is consumed
      __builtin_prefetch((const void*)(Pb + ((h + 1) & 1) * CHUNK + lane), 0, 0);
    }
    __syncthreads();                        // LDS tile visible to all waves

    const int base = h * CHUNK;
    for (int m0 = 0; m0 < CHUNK; m0 += 16) {
      float4 c = sP[m0 + col];              // candidate tile (LDS)
      v2f Aop;
      Aop[0] = half ? c.z : c.x;
      Aop[1] = half ? c.w : c.y;            // c.w = |c|^2
      v8f d = {};
      d = wmma_f32_16x16x4(Aop, Bop, d);
#pragma unroll
      for (int v = 0; v < 8; ++v) {
        float nk = d[v];
        int   ni = base + m0 + rowBase + v;
        if (nk < key[KNN - 1]) {            // guarded sorted insert
          int p = KNN - 1;
#pragma unroll
          for (int j = KNN - 1; j >= 1; --j)
            if (nk < key[j - 1]) { key[j] = key[j - 1]; idx[j] = idx[j - 1]; p = j - 1; }
          key[p] = nk; idx[p] = ni;
        }
      }
    }
    __syncthreads();                        // done reading before next overwrite
  }

  // merge the two half-lane lists per query column (reuse sP as scratch)
  float* mKey = (float*)sP;                 // [8][32][KNN] keys
  int*   mIdx = (int*)sP + 8 * 32 * KNN;    // [8][32][KNN] indices
  const int slot = (wid * 32 + lane) * KNN;
#pragma unroll
  for (int i = 0; i < KNN; ++i) { mKey[slot + i] = key[i]; mIdx[slot + i] = idx[i]; }
  __syncthreads();
  if (lane < 16) {
    const int sa = (wid * 32 + lane) * KNN;
    const int sb = (wid * 32 + lane + 16) * KNN;
    int ia = 0, ib = 0;
    int* out = idxOut + ((size_t)b * NPTS + q0 + lane) * KNN;
    for (int j = 0; j < KNN; ++j) {
      float ka = mKey[sa + ia];
      float kb = mKey[sb + ib];
      bool takeB = kb < ka;
      out[j] = takeB ? mIdx[sb + ib] : mIdx[sa + ia];
      if (takeB) ++ib; else ++ia;
    }
  }
}

// -------- kernel 2: edge features + WMMA GEMM (10->64) + stats + max/min ----
// One wave per point (b,n): 20 edges padded to 32 (pads duplicate edge 0,
// harmless for max/min, masked out of sums). 2 edge tiles x 4 channel tiles
// x 3 K-chunks of WMMA f32 16x16x4.
__global__ __launch_bounds__(256) void edge_kernel(const float* __restrict__ pts,
                                                   const float* __restrict__ W,
                                                   const int* __restrict__ idxBuf,
                                                   float* __restrict__ hmax,
                                                   float* __restrict__ hmin,
                                                   float* __restrict__ stats) {
  __shared__ float sSum[OUTC], sSq[OUTC];
  const int lane = threadIdx.x & 31;
  const int wid  = threadIdx.x >> 5;
  if (threadIdx.x < OUTC) { sSum[threadIdx.x] = 0.f; sSq[threadIdx.x] = 0.f; }
  __syncthreads();

  const int pid  = blockIdx.x * 8 + wid;          // point in [0, BATCH*NPTS)
  const int b    = pid / NPTS, n = pid % NPTS;
  const int col  = lane & 15;
  const int half = lane >> 4;

  // B operands from W (OUTC x INC, row major): 4 channel tiles x 3 K-chunks
  v2f Wop[4][3];
#pragma unroll
  for (int ct = 0; ct < 4; ++ct)
#pragma unroll
    for (int c = 0; c < 3; ++c) {
      int ch = ct * 16 + col;
      int k0 = c * 4 + half * 2;
      Wop[ct][c][0] = (k0     < INC) ? W[ch * INC + k0    ] : 0.f;
      Wop[ct][c][1] = (k0 + 1 < INC) ? W[ch * INC + k0 + 1] : 0.f;
    }

  const int*   nb = idxBuf + ((size_t)b * NPTS + n) * KNN;
  const float* Pb = pts + (size_t)b * NPTS * 5;
  float ci[5];
#pragma unroll
  for (int s = 0; s < 5; ++s) ci[s] = Pb[(size_t)n * 5 + s];

  float sumv[4] = {0, 0, 0, 0}, sqv[4] = {0, 0, 0, 0}, mxv[4], mnv[4];
#pragma unroll
  for (int ct = 0; ct < 4; ++ct) { mxv[ct] = -3.4e38f; mnv[ct] = 3.4e38f; }

  for (int t = 0; t < 2; ++t) {
    int e = t * 16 + col;                         // this lane's edge (A row)
    int g = nb[e < KNN ? e : 0];                  // pads duplicate edge 0
    float cn[5];
#pragma unroll
    for (int s = 0; s < 5; ++s) cn[s] = Pb[(size_t)g * 5 + s];
    float f[12];
    f[0] = cn[0] - ci[0]; f[1] = cn[1] - ci[1]; f[2] = cn[2] - ci[2];
    f[3] = ci[0];         f[4] = ci[1];         f[5] = ci[2];
    f[6] = cn[3] - ci[3]; f[7] = cn[4] - ci[4];
    f[8] = ci[3];         f[9] = ci[4];         f[10] = 0.f; f[11] = 0.f;
    v2f Aop[3];
#pragma unroll
    for (int c = 0; c < 3; ++c) {
      int k0 = c * 4 + half * 2;
      Aop[c][0] = f[k0]; Aop[c][1] = f[k0 + 1];
    }
#pragma unroll
    for (int ct = 0; ct < 4; ++ct) {
      v8f acc = {};
      acc = wmma_f32_16x16x4(Aop[0], Wop[ct][0], acc);
      acc = wmma_f32_16x16x4(Aop[1], Wop[ct][1], acc);
      acc = wmma_f32_16x16x4(Aop[2], Wop[ct][2], acc);
#pragma unroll
      for (int v = 0; v < 8; ++v) {
        float h    = acc[v];
        int   edge = t * 16 + half * 8 + v;
        mxv[ct] = fmaxf(mxv[ct], h);
        mnv[ct] = fminf(mnv[ct], h);
        if (edge < KNN) { sumv[ct] += h; sqv[ct] += h * h; }
      }
    }
  }

  // fold the two half-lanes (same channel, disjoint edges)
#pragma unroll
  for (int ct = 0; ct < 4; ++ct) {
    sumv[ct] += __shfl_xor(sumv[ct], 16, 32);
    sqv[ct]  += __shfl_xor(sqv[ct], 16, 32);
    mxv[ct]   = fmaxf(mxv[ct], __shfl_xor(mxv[ct], 16, 32));
    mnv[ct]   = fminf(mnv[ct], __shfl_xor(mnv[ct], 16, 32));
  }
  if (lane < 16) {
#pragma unroll
    for (int ct = 0; ct < 4; ++ct) {
      int ch = ct * 16 + lane;
      size_t o = ((size_t)b * NPTS + n) * OUTC + ch;
      hmax[o] = mxv[ct];
      hmin[o] = mnv[ct];
      atomicAdd(&sSum[ch], sumv[ct]);
      atomicAdd(&sSq[ch], sqv[ct]);
    }
  }
  __syncthreads();
  if (threadIdx.x < OUTC) {
    atomicAdd(&stats[threadIdx.x], sSum[threadIdx.x]);
    atomicAdd(&stats[OUTC + threadIdx.x], sSq[threadIdx.x]);
  }
}

// -------- kernel 3: finalize BN scale/shift --------
__global__ void finalize_kernel(const float* __restrict__ gamma,
                                const float* __restrict__ beta,
                                float* __restrict__ stats) {
  int ch = threadIdx.x;
  if (ch >= OUTC) return;
  const float cnt  = (float)(BATCH * NPTS * KNN);
  float mean = stats[ch] / cnt;
  float var  = stats[OUTC + ch] / cnt - mean * mean;
  float s    = gamma[ch] * rsqrtf(var + BN_EPS);
  stats[128 + ch] = s;
  stats[192 + ch] = beta[ch] - mean * s;
}

// -------- kernel 4: affine + ReLU + transpose to (B, 64, N) --------
// max over K commutes with monotone relu(affine) when scale>=0 (use hmax),
// and flips to hmin when scale<0.
__global__ __launch_bounds__(256) void out_kernel(const float* __restrict__ hmax,
                                                  const float* __restrict__ hmin,
                                                  const float* __restrict__ stats,
                                                  float* __restrict__ out) {
  size_t i = (size_t)blockIdx.x * 256 + threadIdx.x;   // over B*OUTC*N
  if (i >= (size_t)BATCH * OUTC * NPTS) return;
  int b  = (int)(i / (OUTC * NPTS));
  int r  = (int)(i % (OUTC * NPTS));
  int ch = r / NPTS;
  int n  = r % NPTS;
  float s = stats[128 + ch], t = stats[192 + ch];
  size_t o = ((size_t)b * NPTS + n) * OUTC + ch;
  float v = (s >= 0.f) ? hmax[o] : hmin[o];
  out[i] = fmaxf(s * v + t, 0.f);
}

extern "C" void kernel_launch(void* const* d_in, const int* in_sizes, int n_in,
                              void* d_out, int out_size, void* d_ws, size_t ws_size,
                              hipStream_t stream) {
  const float* pts   = (const float*)d_in[0];
  const float* W     = (const float*)d_in[1];
  const float* gamma = (const float*)d_in[2];
  const float* beta  = (const float*)d_in[3];
  float* ws = (float*)d_ws;

  float4* packed = (float4*)(ws + WS_PACKED);
  int*    idxBuf = (int*)(ws + WS_IDX);
  float*  hmax   = ws + WS_HMAX;
  float*  hmin   = ws + WS_HMIN;
  float*  stats  = ws + WS_STATS;
  float*  out    = (float*)d_out;

  prep_kernel<<<(BATCH * NPTS + 255) / 256, 256, 0, stream>>>(pts, packed, stats);
  knn_kernel<<<BATCH * (NPTS / 128), 256, 0, stream>>>(packed, idxBuf);
  edge_kernel<<<(BATCH * NPTS) / 8, 256, 0, stream>>>(pts, W, idxBuf, hmax, hmin, stats);
  finalize_kernel<<<1, OUTC, 0, stream>>>(gamma, beta, stats);
  out_kernel<<<(unsigned)(((size_t)BATCH * OUTC * NPTS + 255) / 256), 256, 0, stream>>>(
      hmax, hmin, stats, out);
}